// TreeAggregation_27376121544839
// MI455X (gfx1250) — compile-verified
//
#include <hip/hip_runtime.h>

// Problem constants (match reference.py)
#define BB 512       // batch rows of x
#define NN 131072    // columns of x

// ---------------------------------------------------------------------------
// LDS-tiled transpose: src is R x C (row-major), dst is C x R (row-major).
// 32x32 tiles, 33-wide LDS rows to kill bank conflicts; both global sides
// are fully coalesced b32 accesses.
// ---------------------------------------------------------------------------
__global__ __launch_bounds__(256) void tree_transpose_k(
    float* __restrict__ dst, const float* __restrict__ src, int R, int C) {
  __shared__ float tile[32][33];
  int cx = blockIdx.x * 32 + threadIdx.x;   // src col
  int ry = blockIdx.y * 32 + threadIdx.y;   // src row (base)
#pragma unroll
  for (int j = 0; j < 32; j += 8)
    tile[threadIdx.y + j][threadIdx.x] = src[(size_t)(ry + j) * C + cx];
  __syncthreads();
  int dx = blockIdx.y * 32 + threadIdx.x;   // dst col (in [0,R))
  int dy = blockIdx.x * 32 + threadIdx.y;   // dst row (in [0,C))
#pragma unroll
  for (int j = 0; j < 32; j += 8)
    dst[(size_t)(dy + j) * R + dx] = tile[threadIdx.x][threadIdx.y + j];
}

// ---------------------------------------------------------------------------
// Vectorized buffer copy (next := cur) — b128 loads/stores.
// ---------------------------------------------------------------------------
__global__ __launch_bounds__(256) void tree_copy_k(
    float4* __restrict__ dst, const float4* __restrict__ src, int n4) {
  int i = blockIdx.x * 256 + threadIdx.x;
  if (i < n4) dst[i] = src[i];
}

// ---------------------------------------------------------------------------
// Native CDNA5 f32 global atomic add, no-return form (VGLOBAL opcode 86).
// 64-bit VGPR address pair + data VGPR; tracked on STOREcnt; the implicit
// S_WAIT_IDLE at S_ENDPGM guarantees completion before wave retirement.
// ---------------------------------------------------------------------------
__device__ __forceinline__ void atomic_add_f32_native(float* p, float v) {
  asm volatile("global_atomic_add_f32 %0, %1, off"
               :
               : "v"(p), "v"(v)
               : "memory");
}

// ---------------------------------------------------------------------------
// Scatter-add one level in transposed (N x B) layout.
// One wave32 per edge: column = 512 contiguous floats (2 KB).
// Pass p: lane loads float4 at float-offset p*128 + lane*4 -> each
// global_load_b128 across the wave covers a contiguous 512B span.
// Then 4 native f32 atomic adds per pass into the dst column.
// Edge indices are made provably wave-uniform via readfirstlane so the
// int64 index fetches can go down the scalar (s_load_b64 / KMcnt) path.
// ---------------------------------------------------------------------------
__global__ __launch_bounds__(256) void tree_scatter_k(
    float* __restrict__ next, const float* __restrict__ cur,
    const long long* __restrict__ srcIdx, const long long* __restrict__ dstIdx,
    int E) {
  int wave = (int)((blockIdx.x * 256u + threadIdx.x) >> 5);
  int lane = threadIdx.x & 31;
  if (wave >= E) return;

  int wedge = __builtin_amdgcn_readfirstlane(wave);
  int s = (int)srcIdx[wedge];
  int d = (int)dstIdx[wedge];

  const float4* g = (const float4*)(cur + (size_t)s * BB);
  float*        o = next + (size_t)d * BB;

#pragma unroll
  for (int p = 0; p < 4; ++p) {
    float4 v = g[p * 32 + lane];
    int off = (p * 32 + lane) * 4;
    atomic_add_f32_native(o + off + 0, v.x);
    atomic_add_f32_native(o + off + 1, v.y);
    atomic_add_f32_native(o + off + 2, v.z);
    atomic_add_f32_native(o + off + 3, v.w);
  }
}

extern "C" void kernel_launch(void* const* d_in, const int* in_sizes, int n_in,
                              void* d_out, int out_size, void* d_ws, size_t ws_size,
                              hipStream_t stream) {
  const float* x = (const float*)d_in[0];
  const long long* e0 = (const long long*)d_in[1];
  const long long* e1 = (const long long*)d_in[2];
  const long long* e2 = (const long long*)d_in[3];
  const long long* e3 = (const long long*)d_in[4];
  float* out = (float*)d_out;

  const int E  = in_sizes[1] / 2;            // edges are (2, E) int64: [src row; dst row]
  const int n4 = (BB * NN) / 4;              // float4 count of one buffer

  // Ping-pong buffers in transposed (N x B) layout.
  // P0 lives in workspace (256 MB); P1 reuses d_out as scratch — the final
  // transpose overwrites d_out with the correctly-laid-out result.
  float* P0 = (float*)d_ws;
  float* P1 = out;

  // 1) transpose x (B x N) -> P0 (N x B)
  tree_transpose_k<<<dim3(NN / 32, BB / 32), dim3(32, 8), 0, stream>>>(P0, x, BB, NN);

  // 2) four levels, processed edges3 -> edges0 (reference loop order)
  const long long* order[4] = { e3, e2, e1, e0 };
  float* cur = P0;
  float* nxt = P1;
  for (int l = 0; l < 4; ++l) {
    tree_copy_k<<<n4 / 256, 256, 0, stream>>>((float4*)nxt, (const float4*)cur, n4);
    tree_scatter_k<<<(E + 7) / 8, 256, 0, stream>>>(nxt, cur, order[l], order[l] + E, E);
    float* t = cur; cur = nxt; nxt = t;
  }
  // after 4 swaps, cur == P0 (workspace) holds the final state in (N x B)

  // 3) transpose back: cur (N x B) -> d_out (B x N)
  tree_transpose_k<<<dim3(BB / 32, NN / 32), dim3(32, 8), 0, stream>>>(out, cur, NN, BB);
}